// GSULayer_11974368821322
// MI455X (gfx1250) — compile-verified
//
#include <hip/hip_runtime.h>

typedef __attribute__((ext_vector_type(2))) float v2f;
typedef __attribute__((ext_vector_type(8))) float v8f;

#define BN 1024
#define LN 2048
#define EV 16

// ---------------------------------------------------------------------------
// Kernel 0: zero the dice batch-stat accumulators (graph-replay safe).
// ---------------------------------------------------------------------------
__global__ void zero_f32(float* p, int n) {
    int i = blockIdx.x * blockDim.x + threadIdx.x;
    if (i < n) p[i] = 0.0f;
}

// ---------------------------------------------------------------------------
// Kernel 1: fused embedding gather + attention pooling + X_series/mask store.
// One block per batch row b; 256 threads stride over L.
// Writes: xs_out [B,L,48], mask_out [B,L], Xcat [B,96] (= concat(item,pooled))
// ---------------------------------------------------------------------------
__global__ __launch_bounds__(256) void gather_attend(
    const int* __restrict__ ig, const int* __restrict__ ish, const int* __restrict__ ic,
    const int* __restrict__ vg, const int* __restrict__ vsh, const int* __restrict__ vc,
    const float* __restrict__ emb,
    float* __restrict__ xs_out, float* __restrict__ mask_out, float* __restrict__ Xcat)
{
    const int b   = blockIdx.x;
    const int tid = threadIdx.x;

    __shared__ float x_item[48];
    __shared__ float pooled[48];

    if (tid < 48) {
        int f = tid >> 4, e = tid & 15;
        int id = (f == 0) ? ig[b] : (f == 1) ? ish[b] : ic[b];
        x_item[tid] = emb[(size_t)id * EV + e];
        pooled[tid] = 0.0f;
    }
    __syncthreads();

    // hoist item embedding into registers
    float xi[48];
#pragma unroll
    for (int e = 0; e < 48; ++e) xi[e] = x_item[e];

    float acc[48];
#pragma unroll
    for (int e = 0; e < 48; ++e) acc[e] = 0.0f;

    const size_t base = (size_t)b * LN;
    for (int l = tid; l < LN; l += 256) {
        int g = vg[base + l];
        int s = vsh[base + l];
        int c = vc[base + l];
        float4 r[12];
        const float4* pg = (const float4*)(emb + (size_t)g * EV);
        const float4* ps = (const float4*)(emb + (size_t)s * EV);
        const float4* pc = (const float4*)(emb + (size_t)c * EV);
#pragma unroll
        for (int q = 0; q < 4; ++q) { r[q] = pg[q]; r[4 + q] = ps[q]; r[8 + q] = pc[q]; }
        const float* row = (const float*)r;

        float dot = 0.0f;
#pragma unroll
        for (int e = 0; e < 48; ++e) dot += xi[e] * row[e];

        float mk = (g != 0) ? 1.0f : 0.0f;
        float ms = dot * mk;

        float4* outp = (float4*)(xs_out + (base + (size_t)l) * 48);
#pragma unroll
        for (int q = 0; q < 12; ++q) outp[q] = r[q];
        mask_out[base + l] = mk;

#pragma unroll
        for (int e = 0; e < 48; ++e) acc[e] += ms * row[e];
    }
    __syncthreads();
#pragma unroll
    for (int e = 0; e < 48; ++e) atomicAdd(&pooled[e], acc[e]);
    __syncthreads();

    if (tid < 48) {
        Xcat[(size_t)b * 96 + tid]      = x_item[tid];
        Xcat[(size_t)b * 96 + 48 + tid] = pooled[tid];
    }
}

// ---------------------------------------------------------------------------
// Kernel 2: fp32 WMMA GEMM  C[M,N] = A[M,K] @ B[K,N] + bias[N]
// One wave per 16x16 output tile; K is a template constant so the whole
// K-loop fully unrolls into K/4 back-to-back V_WMMA_F32_16X16X4_F32 ops.
// Out-of-range N lanes use a clamped (in-bounds) address instead of a
// predicated load: their accumulator columns are garbage but never stored.
// ---------------------------------------------------------------------------
template <int K>
__global__ __launch_bounds__(32) void gemm_wmma_f32(
    const float* __restrict__ A, const float* __restrict__ Bm,
    const float* __restrict__ bias, float* __restrict__ C,
    int M, int N)
{
    const int tile_m = blockIdx.x * 16;
    const int tile_n = blockIdx.y * 16;
    const int lane   = threadIdx.x;
    const int half   = lane >> 4;   // 0: lanes 0-15, 1: lanes 16-31
    const int l16    = lane & 15;

    const int m    = tile_m + l16;          // A-fragment row for this lane
    const int n    = tile_n + l16;          // B-fragment col for this lane
    const bool nok = (n < N);
    const int  nc  = nok ? n : (N - 1);     // safe in-bounds address

    const float* Ap = A + (size_t)m * K;

    v8f acc = {};
#pragma unroll
    for (int k0 = 0; k0 < K; k0 += 4) {
        const int ka = k0 + 2 * half;       // this lane covers K = ka, ka+1
        v2f a, bfrag;
        a.x = Ap[ka];
        a.y = Ap[ka + 1];
        bfrag.x = Bm[(size_t)ka * N + nc];
        bfrag.y = Bm[(size_t)(ka + 1) * N + nc];
        acc = __builtin_amdgcn_wmma_f32_16x16x4_f32(
            /*neg_a=*/false, a, /*neg_b=*/false, bfrag,
            /*c_mod=*/(short)0, acc, /*reuse_a=*/false, /*reuse_b=*/false);
    }
    // D layout: VGPR r -> M = tile_m + r + 8*half, N = tile_n + (lane&15)
    if (nok) {
        const float bv = bias[n];
#pragma unroll
        for (int r = 0; r < 8; ++r) {
            int mo = tile_m + r + 8 * half;
            C[(size_t)mo * N + n] = acc[r] + bv;
        }
    }
}

// ---------------------------------------------------------------------------
// Kernel 3: per-row LayerNorm (in place) + per-column sum/sumsq accumulation
// (batch statistics for the following Dice). One block per row.
// ---------------------------------------------------------------------------
__global__ __launch_bounds__(256) void layernorm_stats(
    float* __restrict__ H, const float* __restrict__ g, const float* __restrict__ beta,
    float* __restrict__ stats, int N)
{
    const int row = blockIdx.x;
    const int tid = threadIdx.x;
    float* x = H + (size_t)row * N;

    float s = 0.0f, s2 = 0.0f;
    for (int i = tid; i < N; i += 256) { float v = x[i]; s += v; s2 += v * v; }

    __shared__ float rs[256], rs2[256];
    rs[tid] = s; rs2[tid] = s2;
    __syncthreads();
    for (int off = 128; off > 0; off >>= 1) {
        if (tid < off) { rs[tid] += rs[tid + off]; rs2[tid] += rs2[tid + off]; }
        __syncthreads();
    }
    const float m   = rs[0] / (float)N;
    const float var = rs2[0] / (float)N - m * m;
    const float inv = rsqrtf(var + 1e-3f);

    for (int i = tid; i < N; i += 256) {
        float a = g[i] * (x[i] - m) * inv + beta[i];
        x[i] = a;
        atomicAdd(&stats[i],     a);
        atomicAdd(&stats[N + i], a * a);
    }
}

// ---------------------------------------------------------------------------
// Kernel 4: Dice gate (batch-norm stats over B), elementwise in place.
// ---------------------------------------------------------------------------
__global__ void dice_apply(float* __restrict__ H, const float* __restrict__ alpha,
                           const float* __restrict__ stats, int N, int Btot)
{
    int idx = blockIdx.x * blockDim.x + threadIdx.x;
    if (idx >= Btot * N) return;
    int col = idx % N;
    float x  = H[idx];
    float m  = stats[col] / (float)Btot;
    float v  = stats[N + col] / (float)Btot - m * m;
    float xn = (x - m) * rsqrtf(v + 1e-3f);
    float p  = 1.0f / (1.0f + __expf(-xn));
    H[idx] = alpha[col] * (1.0f - p) * x + p * x;
}

// ---------------------------------------------------------------------------
// Kernel 5: final 80->2 projection + softmax. One thread per row.
// ---------------------------------------------------------------------------
__global__ void final_softmax(const float* __restrict__ H2, const float* __restrict__ W3,
                              const float* __restrict__ b3, float* __restrict__ out, int Btot)
{
    int row = blockIdx.x * blockDim.x + threadIdx.x;
    if (row >= Btot) return;
    const float* h = H2 + (size_t)row * 80;
    float z0 = b3[0], z1 = b3[1];
    for (int k = 0; k < 80; ++k) {
        float v = h[k];
        z0 += v * W3[k * 2 + 0];
        z1 += v * W3[k * 2 + 1];
    }
    float mx = fmaxf(z0, z1);
    float e0 = __expf(z0 - mx), e1 = __expf(z1 - mx);
    float s  = 1.0f / (e0 + e1);
    out[row * 2 + 0] = e0 * s;
    out[row * 2 + 1] = e1 * s;
}

// ---------------------------------------------------------------------------
// Launcher
// ---------------------------------------------------------------------------
extern "C" void kernel_launch(void* const* d_in, const int* in_sizes, int n_in,
                              void* d_out, int out_size, void* d_ws, size_t ws_size,
                              hipStream_t stream)
{
    const int*   ig   = (const int*)d_in[0];
    const int*   ish  = (const int*)d_in[1];
    const int*   ic   = (const int*)d_in[2];
    const int*   vg   = (const int*)d_in[3];
    const int*   vsh  = (const int*)d_in[4];
    const int*   vc   = (const int*)d_in[5];
    const float* emb  = (const float*)d_in[6];
    const float* W1   = (const float*)d_in[7];
    const float* b1   = (const float*)d_in[8];
    const float* g1p  = (const float*)d_in[9];
    const float* be1  = (const float*)d_in[10];
    const float* al1  = (const float*)d_in[11];
    const float* W2   = (const float*)d_in[12];
    const float* b2   = (const float*)d_in[13];
    const float* g2p  = (const float*)d_in[14];
    const float* be2  = (const float*)d_in[15];
    const float* al2  = (const float*)d_in[16];
    const float* W3   = (const float*)d_in[17];
    const float* b3   = (const float*)d_in[18];

    float* out      = (float*)d_out;
    float* out_prob = out;                                  // [B,2]
    float* out_xs   = out + 2048;                           // [B,L,48]
    float* out_mask = out + 2048 + (size_t)BN * LN * 48;    // [B,L]

    float* ws = (float*)d_ws;
    float* X  = ws;                 // B*96   = 98304
    float* H1 = X  + 98304;         // B*200  = 204800
    float* H2 = H1 + 204800;        // B*80   = 81920
    float* S1 = H2 + 81920;         // 2*200  = 400
    float* S2 = S1 + 400;           // 2*80   = 160

    // 0) zero dice stats (560 floats)
    zero_f32<<<3, 256, 0, stream>>>(S1, 560);

    // 1) gather + attention + X_series/mask stores (HBM-store bound, ~402 MB)
    gather_attend<<<BN, 256, 0, stream>>>(ig, ish, ic, vg, vsh, vc, emb,
                                          out_xs, out_mask, X);

    // 2) layer 1: X[1024,96] @ W1[96,200] + b1  (WMMA fp32, K=96 -> 24 wmma)
    {
        dim3 grid(BN / 16, (200 + 15) / 16);
        gemm_wmma_f32<96><<<grid, 32, 0, stream>>>(X, W1, b1, H1, BN, 200);
    }
    layernorm_stats<<<BN, 256, 0, stream>>>(H1, g1p, be1, S1, 200);
    dice_apply<<<(BN * 200 + 255) / 256, 256, 0, stream>>>(H1, al1, S1, 200, BN);

    // 3) layer 2: H1[1024,200] @ W2[200,80] + b2  (WMMA fp32, K=200 -> 50 wmma)
    {
        dim3 grid(BN / 16, (80 + 15) / 16);
        gemm_wmma_f32<200><<<grid, 32, 0, stream>>>(H1, W2, b2, H2, BN, 80);
    }
    layernorm_stats<<<BN, 256, 0, stream>>>(H2, g2p, be2, S2, 80);
    dice_apply<<<(BN * 80 + 255) / 256, 256, 0, stream>>>(H2, al2, S2, 80, BN);

    // 4) final projection + softmax
    final_softmax<<<(BN + 255) / 256, 256, 0, stream>>>(H2, W3, b3, out_prob, BN);
}